// Degrade_76235669504050
// MI455X (gfx1250) — compile-verified
//
#include <hip/hip_runtime.h>

typedef float v2f __attribute__((ext_vector_type(2)));
typedef float v8f __attribute__((ext_vector_type(8)));

#define TILE_ROWS  73    // input rows staged per block  (4*15 + 12 + 1)
#define TILE_COLS  140   // input cols staged per block  (64 + 75 + 1)
#define LDS_STRIDE 141   // odd stride -> conflict-free 4-row-strided A loads
#define WPS        136   // padded banded-weight row: t in [-60, 75], +60 bias

// grid: (8, 16, 32)  block: 64 threads (2 waves; each wave one 16x16 out tile)
//   blockIdx.x -> output col tile of 32 (2 waves x 16)
//   blockIdx.y -> output row tile of 16
//   blockIdx.z -> (b*4 + c) plane
__global__ __launch_bounds__(64)
void degrade_wmma_f32_kernel(const float* __restrict__ im,
                             const float* __restrict__ ker,
                             float* __restrict__ out)
{
    __shared__ float S[TILE_ROWS * LDS_STRIDE];
    __shared__ float WP[13 * WPS];

    const int tid  = threadIdx.x;
    const int lane = tid & 31;
    const int wv   = tid >> 5;            // wave id 0..1
    const int bc   = blockIdx.z;          // plane 0..31
    const int b    = bc >> 2;             // batch index (kernel shared by 4 chans)
    const int OY   = blockIdx.y * 16;     // output row base
    const int OXB  = blockIdx.x * 32;     // output col base of block

    // ---- stage replicate-padded input tile into LDS ----
    const long plane_off = (long)bc * (1024L * 1024L);
    const int r0 = 4 * OY  - 6;
    const int c0 = 4 * OXB - 6;
    for (int i = tid; i < TILE_ROWS * TILE_COLS; i += 64) {
        int r = i / TILE_COLS;
        int c = i - r * TILE_COLS;
        int gr = r0 + r; gr = gr < 0 ? 0 : (gr > 1023 ? 1023 : gr);
        int gc = c0 + c; gc = gc < 0 ? 0 : (gc > 1023 ? 1023 : gc);
        S[r * LDS_STRIDE + c] = im[plane_off + (long)gr * 1024 + gc];
    }
    // ---- expand zero-padded banded weights: WP[ky][60+t] = w[ky][t] ----
    for (int i = tid; i < 13 * WPS; i += 64) {
        int ky = i / WPS;
        int t  = i - ky * WPS - 60;
        WP[i] = ((unsigned)t <= 12u) ? ker[b * 169 + ky * 13 + t] : 0.0f;
    }
    __syncthreads();

    // ---- 16x16 output tile via V_WMMA_F32_16X16X4_F32 (fp32-exact) ----
    // A layout (32b 16x4): lanes 0-15 M=0..15 {K0,K1}; lanes 16-31 {K2,K3}
    // B layout (32b 4x16): lanes 0-15 N=0..15 {K0,K1}; lanes 16-31 {K2,K3}
    const int m    = lane & 15;           // A row == B column index n
    const int half = lane >> 4;           // selects K pair {0,1} vs {2,3}

    const int aidx0 = 4 * m * LDS_STRIDE + 64 * wv + 2 * half; // + ky*141 + 4*jc
    const int bidx0 = 60 + 2 * half - 4 * m;                   // + ky*136 + 4*jc

    v8f acc = {};
    for (int jc = 0; jc < 19; ++jc) {     // K chunks of 4 over k = 0..75
        int ai = aidx0 + 4 * jc;
        int bi = bidx0 + 4 * jc;
        #pragma unroll
        for (int ky = 0; ky < 13; ++ky) {
            v2f a;  a.x  = S[ai];      a.y  = S[ai + 1];
            v2f bb; bb.x = WP[bi];     bb.y = WP[bi + 1];
            acc = __builtin_amdgcn_wmma_f32_16x16x4_f32(
                false, a, false, bb, (short)0, acc, false, false);
            ai += LDS_STRIDE;
            bi += WPS;
        }
    }

    // ---- store D: VGPR v -> out row OY + v + 8*half, col = n ----
    const int ocol = OXB + 16 * wv + m;
    float* obase = out + (long)bc * (256L * 256L) + ocol;
    #pragma unroll
    for (int v = 0; v < 8; ++v) {
        const int orow = OY + v + 8 * half;
        obase[(long)orow * 256] = acc[v];
    }
}

extern "C" void kernel_launch(void* const* d_in, const int* in_sizes, int n_in,
                              void* d_out, int out_size, void* d_ws, size_t ws_size,
                              hipStream_t stream) {
    const float* im  = (const float*)d_in[0];   // (8,4,1024,1024) fp32
    const float* ker = (const float*)d_in[1];   // (8,1,13,13) fp32
    float* out = (float*)d_out;                 // (8,4,256,256) fp32
    dim3 grid(8, 16, 32);
    degrade_wmma_f32_kernel<<<grid, dim3(64), 0, stream>>>(im, ker, out);
}